// KVPlugin_17669495455888
// MI455X (gfx1250) — compile-verified
//
#include <hip/hip_runtime.h>
#include <cstdint>
#include <cstddef>

typedef __attribute__((ext_vector_type(4))) float v4f;
typedef __attribute__((ext_vector_type(4))) int   v4i;

#define GLOBAL_AS __attribute__((address_space(1)))
#define LDS_AS    __attribute__((address_space(3)))

namespace {
constexpr int    D               = 128;                      // head dim
constexpr int    ROWS_PER_ARRAY  = 2 * 32 * 2048;            // 131072 = 2^17
constexpr int    ROW_MASK        = ROWS_PER_ARRAY - 1;
constexpr int    S_OLD           = 2048;
constexpr int    S_FULL          = 4096;
constexpr int    CHUNK           = 8;                        // rows per async chunk
constexpr int    CHUNKS_PER_WAVE = 4;
constexpr int    ROWS_PER_WAVE   = CHUNK * CHUNKS_PER_WAVE;  // 32
constexpr int    WAVES_PER_BLOCK = 8;                        // 256 threads
constexpr int    ROWS_PER_BLOCK  = ROWS_PER_WAVE * WAVES_PER_BLOCK; // 256
constexpr int    TOTAL_ROWS      = 4 * ROWS_PER_ARRAY;       // 524288
constexpr int    NBLOCKS         = TOTAL_ROWS / ROWS_PER_BLOCK;     // 2048
constexpr size_t OUT_HALF        = (size_t)2 * 32 * S_FULL * D;     // 33554432 floats
}

// ---- CDNA5 async global->LDS copy (ASYNCcnt-tracked), 16B per lane ----------
__device__ __forceinline__ void async_row_load_b128(const float* gptr, float* lptr) {
#if __has_builtin(__builtin_amdgcn_global_load_async_to_lds_b128)
  __builtin_amdgcn_global_load_async_to_lds_b128(
      (GLOBAL_AS v4i*)gptr, (LDS_AS v4i*)lptr, 0, 0);
#else
  uint32_t laddr = (uint32_t)(uintptr_t)(LDS_AS void*)lptr;
  uint64_t gaddr = (uint64_t)(uintptr_t)gptr;
  asm volatile("global_load_async_to_lds_b128 %0, %1, off"
               :: "v"(laddr), "v"(gaddr)
               : "memory");
#endif
}

#if __has_builtin(__builtin_amdgcn_s_wait_asynccnt)
#define WAIT_ASYNC(n) do { __builtin_amdgcn_s_wait_asynccnt(n); \
                           asm volatile("" ::: "memory"); } while (0)
#else
#define WAIT_ASYNC(n) asm volatile("s_wait_asynccnt " #n ::: "memory")
#endif

// ---- kernel -----------------------------------------------------------------
__global__ __launch_bounds__(256, 1)
void kv_int4_roundtrip(const float* __restrict__ k_cache,
                       const float* __restrict__ v_cache,
                       const float* __restrict__ k_new,
                       const float* __restrict__ v_new,
                       float* __restrict__ out) {
  // double-buffered staging: [wave][buf][row-in-chunk][elem] = 64 KB
  __shared__ __align__(16) float lds[WAVES_PER_BLOCK][2][CHUNK][D];

  const int tid  = threadIdx.x;
  const int lane = tid & 31;
  const int wave = tid >> 5;

  const int blockRow0 = blockIdx.x * ROWS_PER_BLOCK;
  // every block lies fully inside one input array (256 | 2^17)
  const int arr   = blockRow0 >> 17;
  const int isV   = arr & 1;    // 0:K 1:V
  const int isNew = arr >> 1;   // 0:cache 1:new
  const float* base = (arr == 0) ? k_cache
                    : (arr == 1) ? v_cache
                    : (arr == 2) ? k_new
                                 : v_new;

  const int waveRow0 = blockRow0 + wave * ROWS_PER_WAVE;
  const int local0   = waveRow0 & ROW_MASK;   // row index inside its array

  // issue one chunk of 8 async row-copies into buffer `buf`
  auto issue_chunk = [&](int c, int buf) {
    const float* g = base + (size_t)(local0 + c * CHUNK) * D + lane * 4;
#pragma unroll
    for (int r = 0; r < CHUNK; ++r)
      async_row_load_b128(g + r * D, &lds[wave][buf][r][lane * 4]);
  };

  issue_chunk(0, 0);   // prologue

  for (int c = 0; c < CHUNKS_PER_WAVE; ++c) {
    const int buf = c & 1;
    const bool more = (c + 1) < CHUNKS_PER_WAVE;
    if (more) issue_chunk(c + 1, buf ^ 1);
    // wait until the 8 loads of chunk c have landed in LDS
    if (more) WAIT_ASYNC(8); else WAIT_ASYNC(0);

#pragma unroll
    for (int r = 0; r < CHUNK; ++r) {
      v4f x = *(const v4f*)&lds[wave][buf][r][lane * 4];

      // per-row absmax: lane-local then wave32 xor-reduction
      float m = fmaxf(fmaxf(__builtin_fabsf(x.x), __builtin_fabsf(x.y)),
                      fmaxf(__builtin_fabsf(x.z), __builtin_fabsf(x.w)));
      m = fmaxf(m, __shfl_xor(m, 16, 32));
      m = fmaxf(m, __shfl_xor(m,  8, 32));
      m = fmaxf(m, __shfl_xor(m,  4, 32));
      m = fmaxf(m, __shfl_xor(m,  2, 32));
      m = fmaxf(m, __shfl_xor(m,  1, 32));

      const float scale = fmaxf(m / 7.0f, 1e-8f);   // exact IEEE div, like ref

      // quantize -> (pack/unpack is identity on [-7,7]) -> dequantize
      v4f o;
      o.x = fminf(7.0f, fmaxf(-7.0f, __builtin_rintf(x.x / scale))) * scale;
      o.y = fminf(7.0f, fmaxf(-7.0f, __builtin_rintf(x.y / scale))) * scale;
      o.z = fminf(7.0f, fmaxf(-7.0f, __builtin_rintf(x.z / scale))) * scale;
      o.w = fminf(7.0f, fmaxf(-7.0f, __builtin_rintf(x.w / scale))) * scale;

      // output row mapping: [B,H,4096,D], K half then V half, new after old
      const int local  = local0 + c * CHUNK + r;
      const int s      = local & (S_OLD - 1);
      const int bh     = local >> 11;
      const int outRow = bh * S_FULL + isNew * S_OLD + s;
      float* op = out + (size_t)isV * OUT_HALF + (size_t)outRow * D + lane * 4;
      __builtin_nontemporal_store(o, (v4f*)op);   // streaming, write-once
    }
  }
}

// ---- host entry -------------------------------------------------------------
extern "C" void kernel_launch(void* const* d_in, const int* in_sizes, int n_in,
                              void* d_out, int out_size, void* d_ws, size_t ws_size,
                              hipStream_t stream) {
  (void)in_sizes; (void)n_in; (void)out_size; (void)d_ws; (void)ws_size;
  const float* k_cache = (const float*)d_in[0];
  const float* v_cache = (const float*)d_in[1];
  const float* k_new   = (const float*)d_in[2];
  const float* v_new   = (const float*)d_in[3];
  float* out = (float*)d_out;

  kv_int4_roundtrip<<<NBLOCKS, WAVES_PER_BLOCK * 32, 0, stream>>>(
      k_cache, v_cache, k_new, v_new, out);
}